// GraphConvolution2_w_48430051230081
// MI455X (gfx1250) — compile-verified
//
#include <hip/hip_runtime.h>

namespace {

constexpr int N   = 2048;   // nodes
constexpr int F   = 512;    // features
constexpr int MP1 = 11;     // Chebyshev order M+1
constexpr int Q   = 64;     // quadrature points
constexpr float HK_THRESH = 1e-5f;
constexpr int LP  = 18;     // LDS row pitch (floats) -> conflict-free frag reads

typedef __attribute__((ext_vector_type(2))) float v2f;
typedef __attribute__((ext_vector_type(4))) float v4f;
typedef __attribute__((ext_vector_type(8))) float v8f;

// ---------------------------------------------------------------------------
// 1) Per-node Chebyshev coefficients c[i,k] of exp(-exp(t_i)*lam) via
//    Chebyshev-Gauss quadrature.  Also writes t into the output tail.
// ---------------------------------------------------------------------------
__global__ void coeffs_kernel(const float* __restrict__ t,
                              const float* __restrict__ eig,
                              float* __restrict__ C,
                              float* __restrict__ t_out) {
  int i = blockIdx.x * blockDim.x + threadIdx.x;
  if (i >= N) return;
  float ti = t[i];
  float s  = expf(ti);
  float em = eig[0];
  float acc[MP1];
#pragma unroll
  for (int k = 0; k < MP1; ++k) acc[k] = 0.0f;
  for (int q = 0; q < Q; ++q) {
    float theta = 3.14159265358979323846f * (q + 0.5f) / (float)Q;
    float lam   = 0.5f * em * (cosf(theta) + 1.0f);
    float w     = expf(-s * lam);
#pragma unroll
    for (int k = 0; k < MP1; ++k) acc[k] += w * cosf((float)k * theta);
  }
#pragma unroll
  for (int k = 0; k < MP1; ++k) {
    float v = acc[k] * (2.0f / (float)Q);
    if (k == 0) v *= 0.5f;
    C[i * MP1 + k] = v;
  }
  t_out[i] = ti;   // second tuple element of the reference output
}

// ---------------------------------------------------------------------------
// 2) L = (2/eigmax)*P - I ; T0 = I ; T1 = L ; H = c0*I + c1*L
// ---------------------------------------------------------------------------
__global__ void init_kernel(const float* __restrict__ P,
                            const float* __restrict__ eig,
                            const float* __restrict__ C,
                            float* __restrict__ L,
                            float* __restrict__ T0,
                            float* __restrict__ T1,
                            float* __restrict__ H) {
  int idx = blockIdx.x * blockDim.x + threadIdx.x;   // < N*N = 4M
  int row = idx >> 11;            // / 2048
  int col = idx & (N - 1);        // % 2048
  float diag = (row == col) ? 1.0f : 0.0f;
  float l = (2.0f / eig[0]) * P[idx] - diag;
  L[idx]  = l;
  T0[idx] = diag;
  T1[idx] = l;
  H[idx]  = C[row * MP1 + 0] * diag + C[row * MP1 + 1] * l;
}

// ---------------------------------------------------------------------------
// 3) Fused Chebyshev step (LDS-staged WMMA f32 GEMM):
//    acc = L @ Tcur ; Tnext = 2*acc - Tprev ; H += c[:,k] * Tnext
//    Block tile 128x128, 8 waves (4x2), each wave 32x64 (2x4 16x16 tiles).
//    K staged through LDS in chunks of 16; B stored TRANSPOSED so a B
//    fragment (rows ka,ka+1 at one column) is a single ds_load_b64.
// ---------------------------------------------------------------------------
__global__ __launch_bounds__(256)
void cheb_step(const float* __restrict__ L,
               const float* __restrict__ Tcur,
               const float* Tprev,
               float* Tnext,
               float* __restrict__ H,
               const float* __restrict__ C,
               int k) {
  __shared__ float As[128 * LP];   // As[row][kk]  (row-major, pitch LP)
  __shared__ float Bs[128 * LP];   // Bs[col][kk]  (transposed, pitch LP)

  const int tid  = threadIdx.x;
  const int lane = tid & 31;
  const int wave = tid >> 5;
  const int lr   = lane & 15;       // row/col within 16x16 tile
  const int kg   = lane >> 4;       // K-group (0 or 1)

  const int bm = blockIdx.y * 128;
  const int bn = blockIdx.x * 128;
  const int wm = (wave >> 1) * 32;  // 4 waves down
  const int wn = (wave & 1) * 64;   // 2 waves across

  v8f acc[2][4];
#pragma unroll
  for (int m = 0; m < 2; ++m)
#pragma unroll
    for (int n = 0; n < 4; ++n) acc[m][n] = (v8f)0.0f;

  for (int k0 = 0; k0 < N; k0 += 16) {
    // ---- cooperative global loads (all b128) --------------------------
    v4f ar[2], br[2];
#pragma unroll
    for (int r = 0; r < 2; ++r) {
      int ai   = tid + 256 * r;          // 0..511 over 128x16 A chunk
      int arow = ai >> 2;                // 4 float4 per A row
      int ac   = (ai & 3) * 4;
      ar[r] = *(const v4f*)(L + (bm + arow) * N + k0 + ac);
      int brow = ai >> 5;                // 32 float4 per B row (128 cols)
      int bc   = (ai & 31) * 4;
      br[r] = *(const v4f*)(Tcur + (k0 + brow) * N + bn + bc);
    }
    __syncthreads();   // previous chunk fully consumed
#pragma unroll
    for (int r = 0; r < 2; ++r) {
      int ai   = tid + 256 * r;
      int arow = ai >> 2;
      int ac   = (ai & 3) * 4;
      v2f lo = {ar[r][0], ar[r][1]};
      v2f hi = {ar[r][2], ar[r][3]};
      *(v2f*)(As + arow * LP + ac)     = lo;
      *(v2f*)(As + arow * LP + ac + 2) = hi;
      int brow = ai >> 5;
      int bc   = (ai & 31) * 4;
#pragma unroll
      for (int j = 0; j < 4; ++j) Bs[(bc + j) * LP + brow] = br[r][j];
    }
    __syncthreads();   // LDS populated

    // ---- 4 WMMA K-steps from LDS --------------------------------------
#pragma unroll
    for (int ks = 0; ks < 4; ++ks) {
      const int kk = ks * 4 + 2 * kg;    // this lane's K pair within chunk
      v2f a0 = *(const v2f*)(As + (wm + lr) * LP + kk);
      v2f a1 = *(const v2f*)(As + (wm + 16 + lr) * LP + kk);
      v2f b[4];
#pragma unroll
      for (int n = 0; n < 4; ++n)
        b[n] = *(const v2f*)(Bs + (wn + 16 * n + lr) * LP + kk);
#pragma unroll
      for (int n = 0; n < 4; ++n) {
        acc[0][n] = __builtin_amdgcn_wmma_f32_16x16x4_f32(
            false, a0, false, b[n], (short)0, acc[0][n], false, false);
        acc[1][n] = __builtin_amdgcn_wmma_f32_16x16x4_f32(
            false, a1, false, b[n], (short)0, acc[1][n], false, false);
      }
    }
  }

  // Epilogue: Tnext = 2*acc - Tprev ; H += c[row,k] * Tnext
#pragma unroll
  for (int m = 0; m < 2; ++m) {
#pragma unroll
    for (int n = 0; n < 4; ++n) {
#pragma unroll
      for (int v = 0; v < 8; ++v) {
        int row = bm + wm + 16 * m + v + 8 * kg;
        int col = bn + wn + 16 * n + lr;
        int idx = row * N + col;
        float t2 = 2.0f * acc[m][n][v] - Tprev[idx];
        Tnext[idx] = t2;
        H[idx] += C[row * MP1 + k] * t2;
      }
    }
  }
}

// ---------------------------------------------------------------------------
// 4) out = threshold(H) @ x   [2048 x 512], K = 2048.  Same LDS staging;
//    threshold applied once at global load time (before LDS).
// ---------------------------------------------------------------------------
__global__ __launch_bounds__(256)
void hx_gemm(const float* __restrict__ H,
             const float* __restrict__ x,
             float* __restrict__ out) {
  __shared__ float As[128 * LP];
  __shared__ float Bs[128 * LP];

  const int tid  = threadIdx.x;
  const int lane = tid & 31;
  const int wave = tid >> 5;
  const int lr   = lane & 15;
  const int kg   = lane >> 4;

  const int bm = blockIdx.y * 128;
  const int bn = blockIdx.x * 128;
  const int wm = (wave >> 1) * 32;
  const int wn = (wave & 1) * 64;

  v8f acc[2][4];
#pragma unroll
  for (int m = 0; m < 2; ++m)
#pragma unroll
    for (int n = 0; n < 4; ++n) acc[m][n] = (v8f)0.0f;

  for (int k0 = 0; k0 < N; k0 += 16) {
    v4f ar[2], br[2];
#pragma unroll
    for (int r = 0; r < 2; ++r) {
      int ai   = tid + 256 * r;
      int arow = ai >> 2;
      int ac   = (ai & 3) * 4;
      ar[r] = *(const v4f*)(H + (bm + arow) * N + k0 + ac);
#pragma unroll
      for (int j = 0; j < 4; ++j)              // hk[hk < threshold] = 0
        ar[r][j] = (ar[r][j] < HK_THRESH) ? 0.0f : ar[r][j];
      int brow = ai >> 5;
      int bc   = (ai & 31) * 4;
      br[r] = *(const v4f*)(x + (k0 + brow) * F + bn + bc);
    }
    __syncthreads();
#pragma unroll
    for (int r = 0; r < 2; ++r) {
      int ai   = tid + 256 * r;
      int arow = ai >> 2;
      int ac   = (ai & 3) * 4;
      v2f lo = {ar[r][0], ar[r][1]};
      v2f hi = {ar[r][2], ar[r][3]};
      *(v2f*)(As + arow * LP + ac)     = lo;
      *(v2f*)(As + arow * LP + ac + 2) = hi;
      int brow = ai >> 5;
      int bc   = (ai & 31) * 4;
#pragma unroll
      for (int j = 0; j < 4; ++j) Bs[(bc + j) * LP + brow] = br[r][j];
    }
    __syncthreads();

#pragma unroll
    for (int ks = 0; ks < 4; ++ks) {
      const int kk = ks * 4 + 2 * kg;
      v2f a0 = *(const v2f*)(As + (wm + lr) * LP + kk);
      v2f a1 = *(const v2f*)(As + (wm + 16 + lr) * LP + kk);
      v2f b[4];
#pragma unroll
      for (int n = 0; n < 4; ++n)
        b[n] = *(const v2f*)(Bs + (wn + 16 * n + lr) * LP + kk);
#pragma unroll
      for (int n = 0; n < 4; ++n) {
        acc[0][n] = __builtin_amdgcn_wmma_f32_16x16x4_f32(
            false, a0, false, b[n], (short)0, acc[0][n], false, false);
        acc[1][n] = __builtin_amdgcn_wmma_f32_16x16x4_f32(
            false, a1, false, b[n], (short)0, acc[1][n], false, false);
      }
    }
  }

#pragma unroll
  for (int m = 0; m < 2; ++m)
#pragma unroll
    for (int n = 0; n < 4; ++n)
#pragma unroll
      for (int v = 0; v < 8; ++v) {
        int row = bm + wm + 16 * m + v + 8 * kg;
        int col = bn + wn + 16 * n + lr;
        out[row * F + col] = acc[m][n][v];
      }
}

}  // namespace

// ---------------------------------------------------------------------------
// Host-side orchestration (graph-capture safe: only kernel launches on stream)
// ---------------------------------------------------------------------------
extern "C" void kernel_launch(void* const* d_in, const int* in_sizes, int n_in,
                              void* d_out, int out_size, void* d_ws, size_t ws_size,
                              hipStream_t stream) {
  const float* x   = (const float*)d_in[0];  // [N,F]
  const float* P   = (const float*)d_in[1];  // [N,N]
  const float* t   = (const float*)d_in[2];  // [N]
  const float* eig = (const float*)d_in[3];  // [1]
  float* out = (float*)d_out;                // [N*F] output ++ [N] t

  // Workspace layout (floats): C[N*11] | L[N*N] | Tb0[N*N] | Tb1[N*N] | H[N*N]
  float* ws  = (float*)d_ws;
  const size_t NN = (size_t)N * N;
  float* C   = ws;
  float* L   = C + (size_t)N * MP1;   // byte offset divisible by 16 -> v4f ok
  float* Tb0 = L + NN;
  float* Tb1 = Tb0 + NN;
  float* H   = Tb1 + NN;

  // 1) coefficients + copy t to output tail
  coeffs_kernel<<<(N + 255) / 256, 256, 0, stream>>>(t, eig, C, out + (size_t)N * F);

  // 2) L, T0=I, T1=L, H = c0*I + c1*L
  init_kernel<<<(int)(NN / 256), 256, 0, stream>>>(P, eig, C, L, Tb0, Tb1, H);

  // 3) nine fused recurrence GEMMs; Tnext overwrites the Tprev buffer
  float* Ta = Tb0;  // holds T_{k-2}
  float* Tb = Tb1;  // holds T_{k-1}
  for (int k = 2; k <= 10; ++k) {
    cheb_step<<<dim3(N / 128, N / 128), 256, 0, stream>>>(L, Tb, Ta, Ta, H, C, k);
    float* tmp = Ta; Ta = Tb; Tb = tmp;
  }

  // 4) out = threshold(H) @ x
  hx_gemm<<<dim3(F / 128, N / 128), 256, 0, stream>>>(H, x, out);
}